// RerankwMDA_25718264169169
// MI455X (gfx1250) — compile-verified
//
#include <hip/hip_runtime.h>

typedef __attribute__((ext_vector_type(2))) float v2f;
typedef __attribute__((ext_vector_type(8))) float v8f;

#define QN   64
#define MM   400
#define KK   10
#define NTOP 1000
#define DD   512
#define NDB  50000

// One block (256 threads = 8 wave32) per query q.
__global__ __launch_bounds__(256) void rerank_wmma_kernel(
    const int*   __restrict__ rerank_dba_final,   // (Q, M) int32
    const float* __restrict__ res_top1000,        // (Q, M) f32
    const int*   __restrict__ ranks_pre,          // (Q, M) int32 (indices into N_top)
    const float* __restrict__ x_dba,              // (Q, N_top, D) f32
    int*         __restrict__ out)                // (N_db, Q) int32
{
    const int q   = blockIdx.x;
    const int tid = threadIdx.x;

    __shared__ float s_x1max[DD];   // max over first K gathered rows
    __shared__ float s_score[MM];   // dot results, then final scores
    __shared__ float s_sorted[MM];  // descending-sorted res_top values
    __shared__ float s_vals[MM];    // staged res_top values

    const float* xq   = x_dba + (size_t)q * NTOP * DD;
    const int*   idxp = ranks_pre + q * MM;

    // ---------------- Phase A: X1max[d] = max_{k<K} x[q, pre[q,k], d] ----------------
    int idxk[KK];
#pragma unroll
    for (int k = 0; k < KK; ++k) idxk[k] = idxp[k];

    for (int d = tid; d < DD; d += blockDim.x) {
        float mx = -INFINITY;
#pragma unroll
        for (int k = 0; k < KK; ++k)
            mx = fmaxf(mx, xq[(size_t)idxk[k] * DD + d]);
        s_x1max[d] = mx;
    }
    for (int m = tid; m < MM; m += blockDim.x)
        s_vals[m] = res_top1000[q * MM + m];
    __syncthreads();

    // ---------------- Phase B: fp32 WMMA dot products ----------------
    // score_dot[m] = sum_d X2[m,d] * X1max[d], X2[m,:] = x[q, pre[q,m], :]
    // A (16x4 f32): lanes 0-15 hold M=lane, K={0,1}; lanes 16-31 hold M=lane-16, K={2,3}
    // B (4x16 f32): X1max chunk broadcast into all 16 columns (all columns identical)
    const int wave   = tid >> 5;
    const int lane   = tid & 31;
    const int nwaves = blockDim.x >> 5;

    for (int tile = wave; tile < MM / 16; tile += nwaves) {
        const int mrow = tile * 16 + (lane & 15);
        const int koff = (lane < 16) ? 0 : 2;
        const float* xrow = xq + (size_t)idxp[mrow] * DD;

        v8f c = {};
#pragma unroll 4
        for (int s = 0; s < DD / 4; ++s) {
            const int d = s * 4 + koff;                 // even -> 8B aligned
            v2f a = *(const v2f*)(xrow + d);            // A[m, d..d+1]
            v2f b = *(const v2f*)(s_x1max + d);         // B[d..d+1, *] broadcast
            // (neg_a, A, neg_b, B, c_mod, C, reuse_a, reuse_b)
            c = __builtin_amdgcn_wmma_f32_16x16x4_f32(
                    false, a, false, b, (short)0, c, false, false);
        }
        // D layout: VGPR r -> M=r (lanes 0-15), M=r+8 (lanes 16-31); all N equal.
        if (lane == 0) {
#pragma unroll
            for (int r = 0; r < 8; ++r) s_score[tile * 16 + r] = c[r];
        }
        if (lane == 16) {
#pragma unroll
            for (int r = 0; r < 8; ++r) s_score[tile * 16 + 8 + r] = c[r];
        }
    }
    __syncthreads();

    // ---------------- Phase C: stable descending sort of res_top values ----------------
    for (int m = tid; m < MM; m += blockDim.x) {
        const float v = s_vals[m];
        int r = 0;
        for (int j = 0; j < MM; ++j) {
            const float vj = s_vals[j];
            r += (vj > v) || (vj == v && j < m);
        }
        s_sorted[r] = v;   // r is a permutation -> no conflicts
    }
    __syncthreads();

    for (int m = tid; m < MM; m += blockDim.x)
        s_score[m] = (s_sorted[m] + s_score[m]) * 0.5f;
    __syncthreads();

    // ---------------- Phase D: stable descending argsort + scatter ids ----------------
    // rank r of element m => reranked_ids[q, r] = rerank_dba_final[q, m]
    // output[r, q] = reranked_ids[q, r]
    for (int m = tid; m < MM; m += blockDim.x) {
        const float v = s_score[m];
        int r = 0;
        for (int j = 0; j < MM; ++j) {
            const float vj = s_score[j];
            r += (vj > v) || (vj == v && j < m);
        }
        out[(size_t)r * QN + q] = rerank_dba_final[q * MM + m];
    }
}

extern "C" void kernel_launch(void* const* d_in, const int* in_sizes, int n_in,
                              void* d_out, int out_size, void* d_ws, size_t ws_size,
                              hipStream_t stream) {
    (void)in_sizes; (void)n_in; (void)d_ws; (void)ws_size; (void)out_size;

    const int*   ranks            = (const int*)d_in[0];   // (N_db, Q)
    const int*   rerank_dba_final = (const int*)d_in[1];   // (Q, M)
    const float* res_top1000      = (const float*)d_in[2]; // (Q, M)
    const int*   ranks_pre        = (const int*)d_in[3];   // (Q, M)
    const float* x_dba            = (const float*)d_in[4]; // (Q, N_top, D)
    int*         out              = (int*)d_out;           // (N_db, Q)

    // Tail: out[M:, :] = ranks[M:, :]  (contiguous 12.7 MB D2D copy)
    hipMemcpyAsync(out + (size_t)MM * QN,
                   ranks + (size_t)MM * QN,
                   (size_t)(NDB - MM) * QN * sizeof(int),
                   hipMemcpyDeviceToDevice, stream);

    // Head: rerank the first M rows, one block per query.
    rerank_wmma_kernel<<<QN, 256, 0, stream>>>(
        rerank_dba_final, res_top1000, ranks_pre, x_dba, out);
}